// HandmadeConv2DImplementation_7980049236463
// MI455X (gfx1250) — compile-verified
//
#include <hip/hip_runtime.h>

typedef __attribute__((ext_vector_type(2))) float v2f;
typedef __attribute__((ext_vector_type(8))) float v8f;

// Problem constants (from reference)
constexpr int CIN  = 64;
constexpr int COUT = 128;
constexpr int H    = 112;
constexpr int W    = 112;
constexpr int OH   = 110;
constexpr int OW   = 110;
constexpr int KTOT = CIN * 9;            // 576 = GEMM K
constexpr int NT_OW = 32;                // ow positions per block (2 WMMA N-tiles)
constexpr int LDS_ROW   = NT_OW + 2;     // 34 staged input cols per (cin, kh) row
constexpr int LDS_CIN   = 3 * LDS_ROW;   // 102 floats per cin
constexpr int KCHUNK  = 36;              // 4 cins per chunk = 9 WMMA k-steps
constexpr int NCHUNKS = KTOT / KCHUNK;   // 16
constexpr int CIN_PER_WAVE = CIN / 8;    // staging: 8 cins per wave

// Compile-time map: GEMM k -> LDS float offset of im2col element (n = 0 column)
__host__ __device__ constexpr int koff(int kl) {
    return (kl / 9) * LDS_CIN + ((kl % 9) / 3) * LDS_ROW + ((kl % 9) % 3);
}

__launch_bounds__(256)
__global__ void conv2d_implicit_gemm_wmma_f32(const float* __restrict__ in,
                                              const float* __restrict__ wt,
                                              float* __restrict__ out)
{
    __shared__ float smem[CIN * LDS_CIN];

    const int tid     = threadIdx.x;
    const int ow_base = blockIdx.x * NT_OW;
    const int oh      = blockIdx.y;
    const int b       = blockIdx.z;

    const int wave = tid >> 5;
    const int lane = tid & 31;

    // ---- Stage input patch [CIN][3][34] into LDS ----
    // Wave w stages cins [w*8, w*8+8). Lane handles columns idx = lane + 32q.
    // Per-lane (ih, iw) decomposition is done ONCE; the 8-cin inner loops use
    // pure immediate offsets for both the global loads and the LDS stores.
    {
        const float* in_bw = in + ((size_t)b * CIN + wave * CIN_PER_WAVE) * H * W
                                + (size_t)oh * W;
        float* s_w = smem + wave * CIN_PER_WAVE * LDS_CIN;
#pragma unroll
        for (int q = 0; q < 4; ++q) {
            const int idx = lane + 32 * q;            // 0..127; valid < 102
            if (idx < LDS_CIN) {
                const int ih  = idx / LDS_ROW;        // constant-folded divide chain, once
                const int iw  = idx - ih * LDS_ROW;
                const int giw = ow_base + iw;
                const bool ok = (giw < W);
                const int sgiw = ok ? giw : (W - 1);  // clamped, always in-bounds
                const float* g = in_bw + ih * W + sgiw;
                float* s = s_w + idx;
#pragma unroll
                for (int c = 0; c < CIN_PER_WAVE; ++c) {
                    float v = g[c * H * W];           // immediate offset c*12544*4
                    s[c * LDS_CIN] = ok ? v : 0.0f;   // immediate offset c*408
                }
            }
        }
    }
    __syncthreads();

    const int nlo  = lane & 15;   // N position (and A row) within half
    const int half = lane >> 4;   // A/B fragment half: 0 -> K0/K1, 1 -> K2/K3

    // Per-lane A base: this lane supplies W[m, k + 2*half], W[m, k + 2*half + 1]
    const int m = wave * 16 + nlo;
    const float* wlane = wt + (size_t)m * KTOT + 2 * half;

    // One-time half-select of the 18 chunk-local LDS offsets (constants -> VGPRs)
    int offA[9], offB[9];
#pragma unroll
    for (int j = 0; j < 9; ++j) {
        offA[j] = half ? koff(4 * j + 2) : koff(4 * j + 0);
        offB[j] = half ? koff(4 * j + 3) : koff(4 * j + 1);
    }

    const float* sb = smem + nlo;  // per-lane B base, N-tile 0

    v8f acc0 = {};   // ow_base + 0..15
    v8f acc1 = {};   // ow_base + 16..31

    for (int c = 0; c < NCHUNKS; ++c) {
        const float* wk = wlane + c * KCHUNK;       // A chunk base (per lane)
        const float* sc = sb + c * (4 * LDS_CIN);   // B chunk base (4 cins)
#pragma unroll
        for (int j = 0; j < 9; ++j) {
            // A fragment: 8B contiguous, immediate offset from chunk base
            v2f a = *(const v2f*)(wk + 4 * j);

            // B fragments; (b0.x,b1.x) and (b0.y,b1.y) share a base 64B apart
            // -> ds_load_2addr_b32 pairs, conflict-free across lanes
            v2f b0, b1;
            b0.x = sc[offA[j]];
            b0.y = sc[offB[j]];
            b1.x = sc[offA[j] + 16];
            b1.y = sc[offB[j] + 16];

            acc0 = __builtin_amdgcn_wmma_f32_16x16x4_f32(false, a, false, b0,
                                                         (short)0, acc0, false, false);
            acc1 = __builtin_amdgcn_wmma_f32_16x16x4_f32(false, a, false, b1,
                                                         (short)0, acc1, false, false);
        }
    }

    // ---- Store: acc[v] is (cout = wave*16 + v + 8*half, ow = nlo + 16*tile) ----
    // Write-once 198MB stream: non-temporal stores keep it out of L2's way.
    float* pbase = out + (((size_t)b * COUT + wave * 16 + 8 * half) * OH + oh) * OW;

    const int ow0 = ow_base + nlo;
    const int ow1 = ow0 + 16;

    if (ow0 < OW) {
        float* p = pbase + ow0;
#pragma unroll
        for (int v = 0; v < 8; ++v)
            __builtin_nontemporal_store(acc0[v], p + v * (OH * OW));
    }
    if (ow1 < OW) {
        float* p = pbase + ow1;
#pragma unroll
        for (int v = 0; v < 8; ++v)
            __builtin_nontemporal_store(acc1[v], p + v * (OH * OW));
    }
}

extern "C" void kernel_launch(void* const* d_in, const int* in_sizes, int n_in,
                              void* d_out, int out_size, void* d_ws, size_t ws_size,
                              hipStream_t stream) {
    const float* in  = (const float*)d_in[0];   // (32, 64, 112, 112) f32
    const float* wt  = (const float*)d_in[1];   // (128, 64, 3, 3)   f32
    float*       out = (float*)d_out;           // (32, 128, 110, 110) f32

    dim3 grid((OW + NT_OW - 1) / NT_OW, OH, 32);  // (4, 110, 32)
    conv2d_implicit_gemm_wmma_f32<<<grid, 256, 0, stream>>>(in, wt, out);
}